// HeatEquation1D_48962627175209
// MI455X (gfx1250) — compile-verified
//
#include <hip/hip_runtime.h>

typedef __attribute__((ext_vector_type(2))) float v2f;
typedef __attribute__((ext_vector_type(8))) float v8f;

// ---------------------------------------------------------------------------
// Kernel A (tiny, one workgroup): P = (M^T)^100 padded to 64x64, then packed
// into WMMA B-fragment order with a +1 row/col shift so the big GEMM emits
// the zero-padded output directly.
//
// Bw[k][n] = P[k-1][n-1] for 1<=k<=62, 1<=n<=62, else 0.
// Fragment order (matches V_WMMA_F32_16X16X4_F32 B layout, wave32):
//   Pw[((t*16 + kk)*32 + lane)*2 + e] = Bw[kk*4 + (lane>=16?2:0) + e][t*16 + (lane&15)]
// ---------------------------------------------------------------------------
__global__ void __launch_bounds__(256)
heat_matpow_pack(const float* __restrict__ M, float* __restrict__ Pw) {
  __shared__ float Amat[64 * 64];
  __shared__ float cur[64 * 64];
  __shared__ float nxt[64 * 64];
  const int tid = threadIdx.x;

  // A = M^T, zero-padded to 64x64.  (u @ M^T  ==  sum_k u[k] * A[k][j])
  for (int idx = tid; idx < 64 * 64; idx += 256) {
    int i = idx >> 6, j = idx & 63;
    float v = (i < 62 && j < 62) ? M[j * 62 + i] : 0.0f;
    Amat[idx] = v;
    cur[idx] = v;
  }
  __syncthreads();

  // cur = A^100 via 99 sequential 64x64 multiplies (padding is invariant).
  for (int it = 0; it < 99; ++it) {
    for (int idx = tid; idx < 64 * 64; idx += 256) {
      int i = idx >> 6, j = idx & 63;
      float s = 0.0f;
      const float* crow = &cur[i * 64];
#pragma unroll
      for (int k = 0; k < 64; ++k) s = fmaf(crow[k], Amat[k * 64 + j], s);
      nxt[idx] = s;
    }
    __syncthreads();
    for (int idx = tid; idx < 64 * 64; idx += 256) cur[idx] = nxt[idx];
    __syncthreads();
  }

  // Pack shifted matrix into B-fragment order.
  for (int idx = tid; idx < 4 * 16 * 32; idx += 256) {
    int lane = idx & 31;
    int kk = (idx >> 5) & 15;
    int t = idx >> 9;
    int n = t * 16 + (lane & 15);
    int k0 = kk * 4 + ((lane >> 4) << 1);
    float v0 = 0.0f, v1 = 0.0f;
    if (n >= 1 && n <= 62) {
      if (k0 >= 1 && k0 <= 62) v0 = cur[(k0 - 1) * 64 + (n - 1)];
      if (k0 + 1 >= 1 && k0 + 1 <= 62) v1 = cur[(k0 + 0) * 64 + (n - 1)];
    }
    Pw[idx * 2 + 0] = v0;
    Pw[idx * 2 + 1] = v1;
  }
}

// ---------------------------------------------------------------------------
// Kernel B (hot): out[524288 x 64] = u0[524288 x 64] @ Bw[64 x 64]
// One 16-row x 64-col tile per wave32 iteration; B held in 128 VGPRs.
// 64 x v_wmma_f32_16x16x4_f32 per tile. HBM-bound (~11us floor).
// ---------------------------------------------------------------------------
__global__ void __launch_bounds__(256)
heat_gemm(const float* __restrict__ u0, const float* __restrict__ Pw,
          float* __restrict__ out, int numTiles) {
  const int lane = threadIdx.x & 31;
  const int wave = threadIdx.x >> 5;
  const int gwave = blockIdx.x * (blockDim.x >> 5) + wave;
  const int nWaves = gridDim.x * (blockDim.x >> 5);

  // Hoist all B fragments into registers (fully coalesced: 8B/lane).
  v2f b[4][16];
#pragma unroll
  for (int t = 0; t < 4; ++t)
#pragma unroll
    for (int kk = 0; kk < 16; ++kk)
      b[t][kk] = *(const v2f*)(Pw + (((t * 16 + kk) * 32 + lane) << 1));

  const int row = lane & 15;           // A-fragment: M index
  const int colOff = (lane >> 4) << 1; // A-fragment: K sub-offset (0 or 2)
  const int mHalf = (lane >> 4) << 3;  // C/D fragment: row half (0 or 8)
  const int n = lane & 15;             // C/D fragment: N index within tile

  for (int tile = gwave; tile < numTiles; tile += nWaves) {
    const float* ubase = u0 + (size_t)tile * (16 * 64);
    if (tile + nWaves < numTiles)
      __builtin_prefetch(u0 + (size_t)(tile + nWaves) * (16 * 64) + row * 64, 0, 1);

    v8f acc0 = {}, acc1 = {}, acc2 = {}, acc3 = {};
#pragma unroll
    for (int kk = 0; kk < 16; ++kk) {
      v2f a = *(const v2f*)(ubase + row * 64 + kk * 4 + colOff);
      acc0 = __builtin_amdgcn_wmma_f32_16x16x4_f32(false, a, false, b[0][kk],
                                                   (short)0, acc0, false, false);
      acc1 = __builtin_amdgcn_wmma_f32_16x16x4_f32(false, a, false, b[1][kk],
                                                   (short)0, acc1, false, false);
      acc2 = __builtin_amdgcn_wmma_f32_16x16x4_f32(false, a, false, b[2][kk],
                                                   (short)0, acc2, false, false);
      acc3 = __builtin_amdgcn_wmma_f32_16x16x4_f32(false, a, false, b[3][kk],
                                                   (short)0, acc3, false, false);
    }

    float* obase = out + (size_t)tile * (16 * 64);
#pragma unroll
    for (int r = 0; r < 8; ++r) {
      obase[(mHalf + r) * 64 + 0 + n] = acc0[r];
      obase[(mHalf + r) * 64 + 16 + n] = acc1[r];
      obase[(mHalf + r) * 64 + 32 + n] = acc2[r];
      obase[(mHalf + r) * 64 + 48 + n] = acc3[r];
    }
  }
}

extern "C" void kernel_launch(void* const* d_in, const int* in_sizes, int n_in,
                              void* d_out, int out_size, void* d_ws, size_t ws_size,
                              hipStream_t stream) {
  const float* u0 = (const float*)d_in[0];          // [BATCH, 64] f32
  const float* step_matrix = (const float*)d_in[1]; // [62, 62] f32
  float* out = (float*)d_out;                       // [BATCH, 64] f32
  float* Pw = (float*)d_ws;                         // 4096 floats = 16 KB

  const int batch = in_sizes[0] / 64;
  const int numTiles = batch / 16; // 32768 for BATCH=524288

  heat_matpow_pack<<<1, 256, 0, stream>>>(step_matrix, Pw);

  int blocks = 1024;
  if (blocks * 8 > numTiles) blocks = (numTiles + 7) / 8;
  if (blocks < 1) blocks = 1;
  heat_gemm<<<blocks, 256, 0, stream>>>(u0, Pw, out, numTiles);
}